// AdditiveAttention_7816840478957
// MI455X (gfx1250) — compile-verified
//
#include <hip/hip_runtime.h>
#include <hip/hip_bf16.h>

// Problem dims (fixed by the reference)
#define BB 4
#define QQ 512
#define KK 512
#define DD 256
#define HH 256
#define NEGV (-1.0e6f)

typedef float v2f __attribute__((ext_vector_type(2)));
typedef float v8f __attribute__((ext_vector_type(8)));

// ---------------------------------------------------------------------------
// Hardware tanh (gfx1250 TRANS op v_tanh_f32, co-executes with VALU)
// ---------------------------------------------------------------------------
__device__ __forceinline__ float htanh(float x) {
#if __has_builtin(__builtin_amdgcn_tanhf)
  return __builtin_amdgcn_tanhf(x);
#else
  float r;
  asm volatile("v_tanh_f32 %0, %1" : "=v"(r) : "v"(x));
  return r;
#endif
}

// ---------------------------------------------------------------------------
// fp32 WMMA GEMM: C[M,N] = A[M,K] * B[K,N], row-major, batched via blockIdx.y.
// N and K are compile-time so all row strides fold into instruction immediate
// offsets (saddr + 32-bit voffset addressing; single v_add_nc_u32 per iter).
// One 16x32 C tile per wave (8 waves/block), K unrolled by 8 -> 4 wmmas per
// iteration on 4 independent accumulator chains (hides WMMA RAW latency,
// ISA 7.12.1); each A fragment feeds two wmmas.
// Fragment layouts per CDNA5 ISA 7.12.2 (32-bit A/B/C):
//   A 16x4 : lane 0-15 -> M=lane, K={0,1}; lane 16-31 -> M=lane-16, K={2,3}
//   B 4x16 : lane 0-15 -> N=lane, K={0,1}; lane 16-31 -> N=lane-16, K={2,3}
//   C 16x16: VGPR r -> row r + (lane<16 ? 0 : 8), col = lane%16
// Tile counts divide evenly so every wave is fully active (EXEC==all 1s).
// ---------------------------------------------------------------------------
template <int N, int K>
__global__ __launch_bounds__(256) void wmma_gemm_f32_kernel(
    const float* __restrict__ A, const float* __restrict__ Bm,
    float* __restrict__ C, int M, long strideA, long strideB, long strideC) {
  const int batch = blockIdx.y;
  const float* Ab = A + (size_t)batch * (size_t)strideA;   // uniform (SGPR) base
  const float* Bb = Bm + (size_t)batch * (size_t)strideB;  // uniform (SGPR) base
  float* Cb = C + (size_t)batch * (size_t)strideC;

  const int lane = threadIdx.x & 31;
  const int wave = threadIdx.x >> 5;
  constexpr int tilesN = N >> 5;  // 32-wide tiles
  const int tile = blockIdx.x * 8 + wave;
  const int tm = tile / tilesN;
  const int tn = tile % tilesN;

  const int arow = tm * 16 + (lane & 15);
  const int bcol0 = tn * 32 + (lane & 15);
  const int khalf = (lane < 16) ? 0 : 2;

  // 32-bit element offsets against uniform bases -> saddr-form global loads.
  unsigned aoff = (unsigned)arow * K + khalf;
  unsigned boff = (unsigned)khalf * N + bcol0;

  v8f acc00 = {0.f, 0.f, 0.f, 0.f, 0.f, 0.f, 0.f, 0.f};
  v8f acc01 = acc00, acc10 = acc00, acc11 = acc00;

  for (int kk = 0; kk < K; kk += 8) {
    v2f a0, a1, b00, b01, b10, b11;
    a0.x = Ab[aoff + 0];
    a0.y = Ab[aoff + 1];
    a1.x = Ab[aoff + 4];
    a1.y = Ab[aoff + 5];
    b00.x = Bb[boff];
    b00.y = Bb[boff + N];           // constant -> immediate offset
    b10.x = Bb[boff + 16];
    b10.y = Bb[boff + N + 16];
    b01.x = Bb[boff + 4 * N];
    b01.y = Bb[boff + 5 * N];
    b11.x = Bb[boff + 4 * N + 16];
    b11.y = Bb[boff + 5 * N + 16];

    acc00 = __builtin_amdgcn_wmma_f32_16x16x4_f32(false, a0, false, b00,
                                                  (short)0, acc00, false, false);
    acc10 = __builtin_amdgcn_wmma_f32_16x16x4_f32(false, a0, false, b10,
                                                  (short)0, acc10, false, false);
    acc01 = __builtin_amdgcn_wmma_f32_16x16x4_f32(false, a1, false, b01,
                                                  (short)0, acc01, false, false);
    acc11 = __builtin_amdgcn_wmma_f32_16x16x4_f32(false, a1, false, b11,
                                                  (short)0, acc11, false, false);

    aoff += 8;       // single 32-bit constant add
    boff += 8 * N;   // single 32-bit constant add
  }

  const v8f c0 = acc00 + acc01;
  const v8f c1 = acc10 + acc11;
  const int rbase = tm * 16 + ((lane < 16) ? 0 : 8);
  unsigned coff = (unsigned)rbase * N + bcol0;
#pragma unroll
  for (int r = 0; r < 8; ++r) {
    Cb[coff] = c0[r];
    Cb[coff + 16] = c1[r];
    coff += N;
  }
}

// ---------------------------------------------------------------------------
// Additive-attention scores + masked softmax.
// One 256-thread block (8 wave32) per (b,q). Wave w handles key k0+w; lane l
// owns the contiguous slice h in [8*l, 8*l+8): q[h] and wv[h] live in
// registers (loaded once); each k-row costs exactly two global_load_b128 per
// lane against a uniform base + 32-bit offset. 8 x (v_add + v_tanh + v_fma)
// per k per lane; wave32 shuffle reduction.
// ---------------------------------------------------------------------------
__global__ __launch_bounds__(256) void scores_softmax_kernel(
    const float* __restrict__ qp,    // (B*Q, H)
    const float* __restrict__ kp,    // (B*K, H)
    const float* __restrict__ wv,    // (H)
    const int* __restrict__ vlens,   // (B)
    float* __restrict__ attn) {      // (B*Q, K)
  __shared__ float s_sc[KK];
  __shared__ float s_red[16];

  const int bq = blockIdx.x;  // b*Q + q
  const int b = bq >> 9;      // / Q
  const int tid = threadIdx.x;
  const int lane = tid & 31;
  const int wave = tid >> 5;

  // Register-resident q row and wv for this lane's 8 h-values.
  const float4* q4 = (const float4*)(qp + (size_t)bq * HH);
  const float4* w4 = (const float4*)wv;
  const float4 qa = q4[lane * 2 + 0];
  const float4 qb = q4[lane * 2 + 1];
  const float4 wa = w4[lane * 2 + 0];
  const float4 wb = w4[lane * 2 + 1];

  const int vlen = vlens[b];
  const float* kpb = kp + (size_t)b * KK * HH;   // uniform (SGPR) base
  unsigned koff = (unsigned)wave * HH + lane * 8;  // 32-bit element offset

  for (int k0 = 0; k0 < KK; k0 += 8) {
    if (k0 + 8 < KK)  // wave-uniform guard; speculative prefetch of next row
      __builtin_prefetch(kpb + koff + 8 * HH, 0, 0);

    const float4 ka = *(const float4*)(kpb + koff);
    const float4 kb = *(const float4*)(kpb + koff + 4);

    float p = wa.x * htanh(qa.x + ka.x);
    p = fmaf(wa.y, htanh(qa.y + ka.y), p);
    p = fmaf(wa.z, htanh(qa.z + ka.z), p);
    p = fmaf(wa.w, htanh(qa.w + ka.w), p);
    p = fmaf(wb.x, htanh(qb.x + kb.x), p);
    p = fmaf(wb.y, htanh(qb.y + kb.y), p);
    p = fmaf(wb.z, htanh(qb.z + kb.z), p);
    p = fmaf(wb.w, htanh(qb.w + kb.w), p);

#pragma unroll
    for (int off = 16; off > 0; off >>= 1)
      p += __shfl_xor(p, off, 32);
    if (lane == 0) s_sc[k0 + wave] = (k0 + wave < vlen) ? p : NEGV;

    koff += 8 * HH;  // single 32-bit constant add
  }
  __syncthreads();

  // --- masked softmax over K=512 with 256 threads (2 elements each) ---
  const float v0 = s_sc[tid];
  const float v1 = s_sc[tid + 256];
  float m = fmaxf(v0, v1);
#pragma unroll
  for (int off = 16; off > 0; off >>= 1)
    m = fmaxf(m, __shfl_xor(m, off, 32));
  if (lane == 0) s_red[wave] = m;
  __syncthreads();
  float mm = s_red[0];
#pragma unroll
  for (int i = 1; i < 8; ++i) mm = fmaxf(mm, s_red[i]);

  const float e0 = __expf(v0 - mm);
  const float e1 = __expf(v1 - mm);
  float ssum = e0 + e1;
#pragma unroll
  for (int off = 16; off > 0; off >>= 1)
    ssum += __shfl_xor(ssum, off, 32);
  if (lane == 0) s_red[8 + wave] = ssum;  // disjoint slots: no WAR with reads above
  __syncthreads();
  float tot = s_red[8];
#pragma unroll
  for (int i = 1; i < 8; ++i) tot += s_red[8 + i];

  const float inv = 1.0f / tot;
  attn[(size_t)bq * KK + tid] = e0 * inv;
  attn[(size_t)bq * KK + tid + 256] = e1 * inv;
}

// ---------------------------------------------------------------------------
// Launch: qp = queries*Wq ; kp = keys*Wk ; attn = softmax(mask(scores)) ;
//         out = attn @ values  (batched)
// ---------------------------------------------------------------------------
extern "C" void kernel_launch(void* const* d_in, const int* in_sizes, int n_in,
                              void* d_out, int out_size, void* d_ws, size_t ws_size,
                              hipStream_t stream) {
  const float* queries = (const float*)d_in[0];
  const float* keys    = (const float*)d_in[1];
  const float* values  = (const float*)d_in[2];
  const float* Wq      = (const float*)d_in[3];
  const float* Wk      = (const float*)d_in[4];
  const float* wv      = (const float*)d_in[5];
  const int*   vlens   = (const int*)d_in[6];
  float* out = (float*)d_out;

  float* qp   = (float*)d_ws;                  // (B*Q, H)  = 2 MB
  float* kp   = qp + (size_t)BB * QQ * HH;     // (B*K, H)  = 2 MB
  float* attn = kp + (size_t)BB * KK * HH;     // (B*Q, K)  = 4 MB

  dim3 blk(256);

  // Projections: (2048 x 256)*(256 x 256); tiles = 128 x 8 = 1024; /8 waves
  wmma_gemm_f32_kernel<HH, DD><<<dim3(128, 1), blk, 0, stream>>>(
      queries, Wq, qp, BB * QQ, 0, 0, 0);
  wmma_gemm_f32_kernel<HH, DD><<<dim3(128, 1), blk, 0, stream>>>(
      keys, Wk, kp, BB * KK, 0, 0, 0);

  // Scores + masked softmax: one block per (b,q)
  scores_softmax_kernel<<<dim3(BB * QQ), blk, 0, stream>>>(qp, kp, wv, vlens, attn);

  // out[b] = attn[b] (512x512) @ values[b] (512x256); tiles = 32 x 8 = 256; /8
  wmma_gemm_f32_kernel<DD, KK><<<dim3(32, BB), blk, 0, stream>>>(
      attn, values, out, QQ, (long)QQ * KK, (long)KK * DD, (long)QQ * DD);
}